// CrossHeadOnlineHadamardHook_46153718563373
// MI455X (gfx1250) — compile-verified
//
#include <hip/hip_runtime.h>
#include <hip/hip_bf16.h>

// Cross-head Hadamard (size-32 WHT across heads, HEAD_DIM=128) for
// x: (4, 4096, 4096) f32.  Bandwidth-bound (512 MiB traffic -> ~22.5us at
// 23.3 TB/s), so we keep f32 precision and implement the WHT as a GEMM with
// the constant Sylvester H32 (+-1 entries) using V_WMMA_F32_16X16X4_F32.
//
// One 256-thread workgroup (8 wave32) per 4096-float row:
//   phase 1: coalesced 128-bit global loads -> padded LDS
//   phase 2: each wave owns a 16-wide d-slice; 8-step K-chained f32 WMMA
//            (two 16x16 N-tiles), scale by 1/sqrt(32), write back in place
//   phase 3: coalesced 128-bit global stores from LDS

typedef __attribute__((ext_vector_type(2))) float v2f;
typedef __attribute__((ext_vector_type(4))) float v4f;
typedef __attribute__((ext_vector_type(8))) float v8f;

#define HIDDEN      4096
#define NHEAD       32
#define HEAD_DIM    128
#define LDS_STRIDE  129          // pad one float per head-row: kills bank conflicts
#define WHT_SCALE   0.17677669529663688f  // 1/sqrt(32), rounds to the f32 the ref uses

__global__ __launch_bounds__(256) void crosshead_hadamard_wmma(
    const float* __restrict__ x, float* __restrict__ y)
{
    __shared__ float lds[NHEAD * LDS_STRIDE];   // 4128 floats = 16.5 KB

    const int t = threadIdx.x;
    const size_t rowbase = (size_t)blockIdx.x * HIDDEN;
    const float* __restrict__ xr = x + rowbase;
    float* __restrict__ yr = y + rowbase;

    // ---- Phase 1: coalesced load of the row into padded LDS ----
    // 256 threads * 16 floats (4 x float4) = 4096 floats.
    // element e = h*128 + d  ->  lds[h*129 + d] = lds[e + e/128]
#pragma unroll
    for (int j = 0; j < 4; ++j) {
        const int e = t * 16 + j * 4;              // float4 never crosses a head
        const v4f v = *(const v4f*)(xr + e);
        *(v4f*)(&lds[e + (e >> 7)]) = v;
    }
    __syncthreads();

    // ---- Phase 2: per-wave WMMA Hadamard over heads ----
    const int lane = t & 31;
    const int wave = t >> 5;          // 0..7
    const int d0   = wave * 16;       // this wave's exclusive d-slice
    const int m    = lane & 15;       // M (d offset) for A; N column for B/D
    const int hi   = lane >> 4;       // lane-half
    const int koff = hi << 1;         // A/B VGPR0 K offset: 0 or 2
    const int n_lo = m;               // output head for N-tile 0
    const int n_hi = m + 16;          // output head for N-tile 1

    v8f acc0 = {};                    // D tile, heads 0..15
    v8f acc1 = {};                    // D tile, heads 16..31

#pragma unroll
    for (int kk = 0; kk < 8; ++kk) {
        const int k0 = kk * 4 + koff;
        const int k1 = k0 + 1;

        // A fragment (16x4 f32): lane-half h: (M=m, K=k0/k1)
        v2f a;
        a.x = lds[k0 * LDS_STRIDE + d0 + m];
        a.y = lds[k1 * LDS_STRIDE + d0 + m];

        // B fragments (4x16 f32): Sylvester H32[k][n] = (-1)^popc(k&n)
        v2f b0, b1;
        b0.x = (__popc(k0 & n_lo) & 1) ? -1.0f : 1.0f;
        b0.y = (__popc(k1 & n_lo) & 1) ? -1.0f : 1.0f;
        b1.x = (__popc(k0 & n_hi) & 1) ? -1.0f : 1.0f;
        b1.y = (__popc(k1 & n_hi) & 1) ? -1.0f : 1.0f;

        // 8 args: (neg_a, A, neg_b, B, c_mod, C, reuse_a, reuse_b)
        acc0 = __builtin_amdgcn_wmma_f32_16x16x4_f32(
            false, a, false, b0, (short)0, acc0, false, false);
        acc1 = __builtin_amdgcn_wmma_f32_16x16x4_f32(
            false, a, false, b1, (short)0, acc1, false, false);
    }

    // Write D back into LDS in place: D layout vgpr v -> (M = v + 8*hi, N = m).
    // This wave exclusively owns columns d0..d0+15, so in-place is race-free.
#pragma unroll
    for (int v = 0; v < 8; ++v) {
        const int dd = d0 + v + (hi << 3);
        lds[n_lo * LDS_STRIDE + dd] = acc0[v] * WHT_SCALE;
        lds[n_hi * LDS_STRIDE + dd] = acc1[v] * WHT_SCALE;
    }
    __syncthreads();

    // ---- Phase 3: coalesced store of the row from padded LDS ----
#pragma unroll
    for (int j = 0; j < 4; ++j) {
        const int e = t * 16 + j * 4;
        const v4f v = *(const v4f*)(&lds[e + (e >> 7)]);
        *(v4f*)(yr + e) = v;
    }
}

extern "C" void kernel_launch(void* const* d_in, const int* in_sizes, int n_in,
                              void* d_out, int out_size, void* d_ws, size_t ws_size,
                              hipStream_t stream) {
    (void)n_in; (void)out_size; (void)d_ws; (void)ws_size;
    const float* x = (const float*)d_in[0];
    float* y = (float*)d_out;
    const int rows = in_sizes[0] / HIDDEN;   // 4*4096*4096 / 4096 = 16384
    crosshead_hadamard_wmma<<<dim3(rows), dim3(256), 0, stream>>>(x, y);
}